// GCN_24129126268988
// MI455X (gfx1250) — compile-verified
//
#include <hip/hip_runtime.h>
#include <math.h>

#define TF_IN 128
#define THID  16
#define TCLS  4
#define BN_EPS 1e-5f

typedef __attribute__((ext_vector_type(2))) float v2f;
typedef __attribute__((ext_vector_type(8))) float v8f;

// ---------------------------------------------------------------------------
// GEMM1: h1[N,16] = x[N,128] @ W1[128,16] using V_WMMA_F32_16X16X4_F32.
// One wave per 16-row tile. K loop of 32 WMMA steps (K=128, 4 per step).
// A 16x4 layout (ISA 7.12.2): lane l -> row (l&15); VGPR0/1 hold K = base+{0,1}
// for lanes 0-15 and K = base+{2,3} for lanes 16-31. B 4x16 is symmetric.
// D 16x16: VGPR i, lane l -> row (i + 8*(l>=16)), col (l&15).
// ---------------------------------------------------------------------------
__global__ void __launch_bounds__(128)
gcn_gemm1_wmma(const float* __restrict__ x, const float* __restrict__ W,
               float* __restrict__ h, int nTiles, int N) {
  const int gwave = (int)((blockIdx.x * blockDim.x + threadIdx.x) >> 5);
  if (gwave >= nTiles) return;                 // wave-uniform: EXEC stays full
  const int lane = (int)(threadIdx.x & 31);
  const int m  = lane & 15;
  const int hi = lane >> 4;
  int row = gwave * 16 + m;
  if (row > N - 1) row = N - 1;                // clamp tail reads (dup rows ok)
  const float* __restrict__ xr = x + (size_t)row * TF_IN;
  v8f acc = {};
#pragma unroll
  for (int k0 = 0; k0 < TF_IN; k0 += 4) {
    const int kb = k0 + (hi << 1);
    v2f a, b;
    a.x = xr[kb];
    a.y = xr[kb + 1];
    b.x = W[kb * THID + m];
    b.y = W[(kb + 1) * THID + m];
    acc = __builtin_amdgcn_wmma_f32_16x16x4_f32(false, a, false, b,
                                                (short)0, acc, false, false);
  }
#pragma unroll
  for (int i = 0; i < 8; ++i) {
    const int r = gwave * 16 + i + hi * 8;
    if (r < N) h[(size_t)r * THID + m] = acc[i];
  }
}

// ---------------------------------------------------------------------------
// GEMM2: h2[N,4] = h1[N,16] @ W2[16,4]; W2 zero-padded to 16x16 on the fly.
// ---------------------------------------------------------------------------
__global__ void __launch_bounds__(128)
gcn_gemm2_wmma(const float* __restrict__ h, const float* __restrict__ W2,
               float* __restrict__ h2, int nTiles, int N) {
  const int gwave = (int)((blockIdx.x * blockDim.x + threadIdx.x) >> 5);
  if (gwave >= nTiles) return;
  const int lane = (int)(threadIdx.x & 31);
  const int m  = lane & 15;
  const int hi = lane >> 4;
  int row = gwave * 16 + m;
  if (row > N - 1) row = N - 1;
  const float* __restrict__ hr = h + (size_t)row * THID;
  v8f acc = {};
#pragma unroll
  for (int k0 = 0; k0 < THID; k0 += 4) {
    const int kb = k0 + (hi << 1);
    v2f a, b;
    a.x = hr[kb];
    a.y = hr[kb + 1];
    b.x = (m < TCLS) ? W2[kb * TCLS + m] : 0.0f;
    b.y = (m < TCLS) ? W2[(kb + 1) * TCLS + m] : 0.0f;
    acc = __builtin_amdgcn_wmma_f32_16x16x4_f32(false, a, false, b,
                                                (short)0, acc, false, false);
  }
  if (m < TCLS) {
#pragma unroll
    for (int i = 0; i < 8; ++i) {
      const int r = gwave * 16 + i + hi * 8;
      if (r < N) h2[(size_t)r * TCLS + m] = acc[i];
    }
  }
}

// Degree counting: deg[dst] += 1 per edge (deg pre-zeroed).
__global__ void gcn_degree(const int* __restrict__ dst, float* __restrict__ deg,
                           int E) {
  const int e = blockIdx.x * blockDim.x + threadIdx.x;
  if (e < E) atomicAdd(&deg[dst[e]], 1.0f);
}

// dinv = rsqrt(1 + indeg), in place.
__global__ void gcn_dinv(float* __restrict__ deg, int n) {
  const int i = blockIdx.x * blockDim.x + threadIdx.x;
  if (i < n) deg[i] = rsqrtf(deg[i] + 1.0f);
}

// Edge scatter: agg[dst,f] += h[src,f] * dinv[src]*dinv[dst].
// One thread per (edge, feature); F = 1<<lf.
__global__ void gcn_agg(const float* __restrict__ h, const int* __restrict__ src,
                        const int* __restrict__ dst, const float* __restrict__ dinv,
                        float* __restrict__ agg, long long total, int lf) {
  const long long idx = (long long)blockIdx.x * blockDim.x + threadIdx.x;
  if (idx >= total) return;
  const int F = 1 << lf;
  const int e = (int)(idx >> lf);
  const int f = (int)(idx & (F - 1));
  const int s = src[e];
  const int d = dst[e];
  const float nrm = dinv[s] * dinv[d];
  atomicAdd(&agg[(size_t)d * F + f], h[(size_t)s * F + f] * nrm);
}

// Finalize: z = agg + h*dinv^2 + bias (in place on agg), accumulate BN stats
// (sum, sumsq per channel) via LDS atomics then one global atomic per block.
__global__ void gcn_finalize_stats(const float* __restrict__ h,
                                   const float* __restrict__ dinv,
                                   const float* __restrict__ bias,
                                   float* __restrict__ z,
                                   float* __restrict__ stats, int n, int F) {
  __shared__ float ssum[16];
  __shared__ float ssq[16];
  if ((int)threadIdx.x < F) {
    ssum[threadIdx.x] = 0.0f;
    ssq[threadIdx.x] = 0.0f;
  }
  __syncthreads();
  const int i = blockIdx.x * blockDim.x + threadIdx.x;
  if (i < n) {
    const float di = dinv[i];
    const float d2 = di * di;
    for (int f = 0; f < F; ++f) {
      const float v = z[(size_t)i * F + f] + h[(size_t)i * F + f] * d2 + bias[f];
      z[(size_t)i * F + f] = v;
      atomicAdd(&ssum[f], v);
      atomicAdd(&ssq[f], v * v);
    }
  }
  __syncthreads();
  if ((int)threadIdx.x < F) {
    atomicAdd(&stats[threadIdx.x], ssum[threadIdx.x]);
    atomicAdd(&stats[F + threadIdx.x], ssq[threadIdx.x]);
  }
}

// BN (training-mode batch stats) + ReLU, elementwise.
__global__ void gcn_bn_relu(const float* __restrict__ z,
                            const float* __restrict__ stats,
                            const float* __restrict__ g,
                            const float* __restrict__ beta,
                            float* __restrict__ out, long long total, int lf,
                            float invN) {
  const long long idx = (long long)blockIdx.x * blockDim.x + threadIdx.x;
  if (idx >= total) return;
  const int F = 1 << lf;
  const int f = (int)(idx & (F - 1));
  const float m = stats[f] * invN;
  const float v = stats[F + f] * invN - m * m;
  float val = (z[idx] - m) * rsqrtf(v + BN_EPS) * g[f] + beta[f];
  out[idx] = val > 0.0f ? val : 0.0f;
}

// BN + ReLU + log_softmax over the 4 classes, per node.
__global__ void gcn_bn_relu_lsm(const float* __restrict__ z,
                                const float* __restrict__ stats,
                                const float* __restrict__ g,
                                const float* __restrict__ beta,
                                float* __restrict__ out, int n, float invN) {
  const int i = blockIdx.x * blockDim.x + threadIdx.x;
  if (i >= n) return;
  float y[TCLS];
  float mx = -1e30f;
#pragma unroll
  for (int c = 0; c < TCLS; ++c) {
    const float m = stats[c] * invN;
    const float v = stats[TCLS + c] * invN - m * m;
    float val = (z[(size_t)i * TCLS + c] - m) * rsqrtf(v + BN_EPS) * g[c] + beta[c];
    val = val > 0.0f ? val : 0.0f;
    y[c] = val;
    mx = fmaxf(mx, val);
  }
  float s = 0.0f;
#pragma unroll
  for (int c = 0; c < TCLS; ++c) s += __expf(y[c] - mx);
  const float lse = mx + __logf(s);
#pragma unroll
  for (int c = 0; c < TCLS; ++c) out[(size_t)i * TCLS + c] = y[c] - lse;
}

extern "C" void kernel_launch(void* const* d_in, const int* in_sizes, int n_in,
                              void* d_out, int out_size, void* d_ws, size_t ws_size,
                              hipStream_t stream) {
  const float* x   = (const float*)d_in[0];
  const int*   ei  = (const int*)d_in[1];
  const float* W1  = (const float*)d_in[2];
  const float* b1  = (const float*)d_in[3];
  const float* g1  = (const float*)d_in[4];
  const float* be1 = (const float*)d_in[5];
  const float* W2  = (const float*)d_in[6];
  const float* b2  = (const float*)d_in[7];
  const float* g2  = (const float*)d_in[8];
  const float* be2 = (const float*)d_in[9];
  float* out = (float*)d_out;

  const int N = in_sizes[0] / TF_IN;
  const int E = in_sizes[1] / 2;
  const int* src = ei;
  const int* dst = ei + E;

  // Workspace layout (floats). Accumulator region first so one memset clears it.
  float* ws   = (float*)d_ws;
  float* agg1 = ws;                              // N*16
  float* agg2 = agg1 + (size_t)N * THID;         // N*4
  float* deg  = agg2 + (size_t)N * TCLS;         // N  (becomes dinv)
  float* st1  = deg + N;                         // 32 (sum16 | sumsq16)
  float* st2  = st1 + 32;                        // 32 (sum4 | sumsq4, padded)
  float* h1   = st2 + 32;                        // N*16
  float* h2   = h1 + (size_t)N * THID;           // N*4

  const size_t zeroFloats = (size_t)N * (THID + TCLS + 1) + 64;
  hipMemsetAsync(d_ws, 0, zeroFloats * sizeof(float), stream);

  const int nTiles = (N + 15) / 16;
  const float invN = 1.0f / (float)N;

  // Layer 1
  gcn_gemm1_wmma<<<(nTiles + 3) / 4, 128, 0, stream>>>(x, W1, h1, nTiles, N);
  gcn_degree<<<(E + 255) / 256, 256, 0, stream>>>(dst, deg, E);
  gcn_dinv<<<(N + 255) / 256, 256, 0, stream>>>(deg, N);
  {
    const long long total = (long long)E * THID;
    gcn_agg<<<(int)((total + 255) / 256), 256, 0, stream>>>(h1, src, dst, deg,
                                                            agg1, total, 4);
  }
  gcn_finalize_stats<<<(N + 255) / 256, 256, 0, stream>>>(h1, deg, b1, agg1,
                                                          st1, N, THID);
  {
    const long long total = (long long)N * THID;
    gcn_bn_relu<<<(int)((total + 255) / 256), 256, 0, stream>>>(agg1, st1, g1,
                                                                be1, h1, total,
                                                                4, invN);
  }

  // Layer 2
  gcn_gemm2_wmma<<<(nTiles + 3) / 4, 128, 0, stream>>>(h1, W2, h2, nTiles, N);
  {
    const long long total = (long long)E * TCLS;
    gcn_agg<<<(int)((total + 255) / 256), 256, 0, stream>>>(h2, src, dst, deg,
                                                            agg2, total, 2);
  }
  gcn_finalize_stats<<<(N + 255) / 256, 256, 0, stream>>>(h2, deg, b2, agg2,
                                                          st2, N, TCLS);
  gcn_bn_relu_lsm<<<(N + 255) / 256, 256, 0, stream>>>(agg2, st2, g2, be2, out,
                                                       N, invN);
}